// Classifier_4535485464746
// MI455X (gfx1250) — compile-verified
//
#include <hip/hip_runtime.h>

typedef float v2f __attribute__((ext_vector_type(2)));
typedef float v8f __attribute__((ext_vector_type(8)));

#define IMG_DIM 3072
#define NSTATE  128
#define NCLS    10

// ---------------------------------------------------------------------------
// Kernel 1: x0[b][n] = sum_k img[b][k] * w_in[n][k]
// fp32 WMMA 16x16x4, one wave per 16x16 output tile, 8 waves per block.
// A tile: rows of img (M=batch), B tile: rows of w_in (N=state), both
// K-contiguous in memory -> plain b64 loads per lane per K-step.
// ---------------------------------------------------------------------------
__global__ __launch_bounds__(256)
void gemm_x0_wmma(const float* __restrict__ img,
                  const float* __restrict__ w_in,
                  float* __restrict__ x0) {
  const int lane = threadIdx.x & 31;
  const int wave = threadIdx.x >> 5;
  const int tile = blockIdx.x * 8 + wave;   // 512 tiles total
  const int tm = tile >> 3;                 // 64 M-tiles (batch/16)
  const int tn = tile & 7;                  // 8  N-tiles (128/16)

  // A 16x4 f32 layout: lanes 0-15 hold M=lane, v0=K0,v1=K1;
  //                    lanes 16-31 hold M=lane-16, v0=K2,v1=K3.
  // B 4x16 f32 layout mirrors with N in place of M.
  const int r  = lane & 15;
  const int kh = (lane >> 4) * 2;           // K sub-offset 0 or 2

  const float* aP = img  + (size_t)(tm * 16 + r) * IMG_DIM + kh;
  const float* bP = w_in + (size_t)(tn * 16 + r) * IMG_DIM + kh;

  v8f acc = {};
#pragma unroll 8
  for (int k = 0; k < IMG_DIM; k += 4) {
    v2f a = *(const v2f*)(aP + k);
    v2f b = *(const v2f*)(bP + k);
    // 8 args: (neg_a, A, neg_b, B, c_mod, C, reuse_a, reuse_b)
    acc = __builtin_amdgcn_wmma_f32_16x16x4_f32(false, a, false, b,
                                                (short)0, acc, false, false);
  }

  // C/D layout: VGPR i -> (M = i + (lane<16 ? 0 : 8), N = lane&15)
  const int n  = lane & 15;
  const int mb = (lane >> 4) * 8;
#pragma unroll
  for (int i = 0; i < 8; ++i)
    x0[(size_t)(tm * 16 + mb + i) * NSTATE + tn * 16 + n] = acc[i];
}

// ---------------------------------------------------------------------------
// Kernel 2: 512-step Lorenz-96 Euler scan + readout GEMM.
// One wave32 per batch row; lane l owns state elements 4l..4l+3 in registers.
// Cyclic rolls via 3 wave shuffles per step (ds_bpermute), no LDS/barriers.
// Readout: per-lane partials + __shfl_xor butterfly reduction.
// ---------------------------------------------------------------------------
__global__ __launch_bounds__(256)
void lorenz_scan_readout(const float* __restrict__ x0,
                         const float* __restrict__ w_out,
                         const float* __restrict__ forcing_p,
                         const int* __restrict__ nsteps_p,
                         float* __restrict__ out) {
  const int lane = threadIdx.x & 31;
  const int wave = threadIdx.x >> 5;
  const int row  = blockIdx.x * 8 + wave;   // 0..1023

  const float F  = *forcing_p;
  const int   ns = *nsteps_p;
  const float dt = 1.0f / (float)ns;        // READOUT_TIME = 1.0

  const float4 xin = *(const float4*)(x0 + (size_t)row * NSTATE + lane * 4);
  float e0 = xin.x, e1 = xin.y, e2 = xin.z, e3 = xin.w;

  const int laneM1 = (lane + 31) & 31;
  const int laneP1 = (lane + 1) & 31;

  for (int s = 0; s < ns; ++s) {
    // neighbors: x[4l-2], x[4l-1] from lane-1 (its e2,e3); x[4l+4] from lane+1 (its e0)
    float p2 = __shfl(e2, laneM1, 32);
    float p3 = __shfl(e3, laneM1, 32);
    float n0 = __shfl(e0, laneP1, 32);

    // dx_i = (x_{i+1} - x_{i-2}) * x_{i-1} - x_i + F
    float d0 = (e1 - p2) * p3 - e0 + F;
    float d1 = (e2 - p3) * e0 - e1 + F;
    float d2 = (e3 - e0) * e1 - e2 + F;
    float d3 = (n0 - e1) * e2 - e3 + F;

    e0 = fmaf(dt, d0, e0);
    e1 = fmaf(dt, d1, e1);
    e2 = fmaf(dt, d2, e2);
    e3 = fmaf(dt, d3, e3);
  }

  // readout: out[row][c] = sum_n w_out[c][n] * x[n]
  float acc[NCLS];
#pragma unroll
  for (int c = 0; c < NCLS; ++c) {
    const float4 w = *(const float4*)(w_out + (size_t)c * NSTATE + lane * 4);
    acc[c] = w.x * e0 + w.y * e1 + w.z * e2 + w.w * e3;
  }
#pragma unroll
  for (int off = 16; off > 0; off >>= 1) {
#pragma unroll
    for (int c = 0; c < NCLS; ++c)
      acc[c] += __shfl_xor(acc[c], off, 32);
  }
  if (lane == 0) {
#pragma unroll
    for (int c = 0; c < NCLS; ++c)
      out[(size_t)row * NCLS + c] = acc[c];
  }
}

// ---------------------------------------------------------------------------
extern "C" void kernel_launch(void* const* d_in, const int* in_sizes, int n_in,
                              void* d_out, int out_size, void* d_ws, size_t ws_size,
                              hipStream_t stream) {
  const float* img     = (const float*)d_in[0];   // [1024, 3072]
  const float* w_in    = (const float*)d_in[1];   // [128, 3072]
  const float* w_out   = (const float*)d_in[2];   // [10, 128]
  const float* forcing = (const float*)d_in[3];   // scalar
  const int*   n_steps = (const int*)d_in[4];     // scalar

  float* x0  = (float*)d_ws;                      // [1024, 128] scratch
  float* out = (float*)d_out;                     // [1024, 10]

  // 512 WMMA tiles, 8 waves (tiles) per 256-thread block -> 64 blocks
  gemm_x0_wmma<<<64, 256, 0, stream>>>(img, w_in, x0);

  // 1024 rows, 8 waves (rows) per 256-thread block -> 128 blocks
  lorenz_scan_readout<<<128, 256, 0, stream>>>(x0, w_out, forcing, n_steps, out);
}